// VectorQuantizer_58557584113934
// MI455X (gfx1250) — compile-verified
//
#include <hip/hip_runtime.h>

// ---------------------------------------------------------------------------
// VQ-VAE vector quantizer forward for MI455X (gfx1250, wave32).
//
//  z:   [32, 64, 32, 32] f32   -> N = 32768 rows of dim K = 64
//  emb: [1024, 64] f32         -> 1024 codes
//
//  Core: argmin_j (||e_j||^2 - 2 z.e_j) via V_WMMA_F32_16X16X4_F32
//  (f32 matrix path: M=32768, N=1024, K=64). ||z||^2 dropped: constant per
//  row, cannot change the argmin.
//
//  Roofline: dominated by the 128 MB one-hot store (~5.7us @23.3TB/s HBM).
//  Those stores (and the 8 MB z_q) are non-temporal so the 192 MB L2 stays
//  dedicated to emb/z re-reads. GEMM (~4.3 GFLOP) runs dual-accumulator
//  per wave to break the WMMA D->C dependency chain.
//  All reductions are fixed-order (no float atomics) -> deterministic replay.
// ---------------------------------------------------------------------------

typedef __attribute__((ext_vector_type(2))) float v2f;
typedef __attribute__((ext_vector_type(8))) float v8f;

#define NVEC   32768      // 32*32*32 rows
#define EDIM   64
#define NE     1024
#define HW     1024       // 32*32
#define CHW    65536      // 64*1024
#define TOT    2097152    // 32*64*32*32

// ---------------------------------------------------------------------------
// ||e_j||^2 for all 1024 codes (tiny: 256 KB read)
// ---------------------------------------------------------------------------
__global__ void k_enorm(const float* __restrict__ emb, float* __restrict__ enorm) {
    int j = blockIdx.x * blockDim.x + threadIdx.x;
    if (j < NE) {
        float s = 0.f;
        #pragma unroll
        for (int k = 0; k < EDIM; ++k) { float e = emb[j * EDIM + k]; s += e * e; }
        enorm[j] = s;
    }
}

// ---------------------------------------------------------------------------
// GEMM + argmin. One wave per 16-row M-tile; two N-tiles in flight per step
// (independent WMMA accumulator chains c0/c1 sharing the A fragments).
//
// f32 16x16x4 operand layout (ISA 7.12.2):
//   A (16x4): lanes 0-15 -> M=lane, VGPR0/1 = K0/K1; lanes 16-31 -> K2/K3
//   B (4x16): lanes 0-15 -> N=lane, VGPR0/1 = K0/K1; lanes 16-31 -> K2/K3
//   C (16x16): VGPR r -> M=r (lanes 0-15), M=r+8 (lanes 16-31), N=lane&15
// ---------------------------------------------------------------------------
__global__ void __launch_bounds__(128) k_gemm_argmin(
        const float* __restrict__ z,       // NCHW [32,64,32,32]
        const float* __restrict__ emb,     // [1024,64]
        const float* __restrict__ enorm,   // [1024]
        int*   __restrict__ idx_out,       // [NVEC] scratch (int)
        float* __restrict__ idx_f_out)     // [NVEC] d_out indices-as-float
{
    const int lane  = threadIdx.x & 31;
    const int wave  = blockIdx.x * (blockDim.x >> 5) + (threadIdx.x >> 5);
    const int mbase = wave * 16;
    const int ml    = lane & 15;          // row/col within tile
    const int kb    = (lane >> 4) * 2;    // K sub-offset for this half-wave

    // --- preload A fragments for all 16 K-chunks (strided NCHW gather,
    //     coalesced across lanes: rows of a tile share b, consecutive hw) ---
    const int m  = mbase + ml;
    const int b  = m >> 10;
    const int hw = m & 1023;
    const float* zbase = z + b * CHW + hw;
    v2f a[16];
    #pragma unroll
    for (int kc = 0; kc < 16; ++kc) {
        const int k = kc * 4 + kb;
        v2f t; t.x = zbase[k * HW]; t.y = zbase[(k + 1) * HW];
        a[kc] = t;
    }

    float best[8];
    int   bidx[8];
    #pragma unroll
    for (int r = 0; r < 8; ++r) { best[r] = 3.4e38f; bidx[r] = 0; }

    for (int ntp = 0; ntp < 32; ++ntp) {        // 32 pairs of N-tiles
        const int   n0  = ntp * 32 + ml;
        const int   n1  = n0 + 16;
        const float en0 = enorm[n0];
        const float en1 = enorm[n1];
        const float* brow0 = emb + n0 * EDIM + kb;
        const float* brow1 = emb + n1 * EDIM + kb;

        v8f c0 = {};
        v8f c1 = {};
        #pragma unroll
        for (int kc = 0; kc < 16; ++kc) {
            v2f bf0 = *(const v2f*)(brow0 + kc * 4);
            v2f bf1 = *(const v2f*)(brow1 + kc * 4);
            c0 = __builtin_amdgcn_wmma_f32_16x16x4_f32(
                    false, a[kc], false, bf0, (short)0, c0, false, false);
            c1 = __builtin_amdgcn_wmma_f32_16x16x4_f32(
                    false, a[kc], false, bf1, (short)0, c1, false, false);
        }
        #pragma unroll
        for (int r = 0; r < 8; ++r) {
            const float s0 = en0 - 2.0f * c0[r];
            if (s0 < best[r]) { best[r] = s0; bidx[r] = n0; }
            const float s1 = en1 - 2.0f * c1[r];
            if (s1 < best[r]) { best[r] = s1; bidx[r] = n1; }
            // per-lane n grows monotonically (n0 < n1 < next pair) ->
            // strict '<' keeps the first index on ties, like argmin
        }
    }

    // --- argmin across the 16 lanes of each half (xor of bits 0..3 stays
    //     inside the half); tie-break toward the smaller index like argmin ---
    #pragma unroll
    for (int r = 0; r < 8; ++r) {
        float s = best[r];
        int   i = bidx[r];
        #pragma unroll
        for (int msk = 8; msk >= 1; msk >>= 1) {
            const float os = __shfl_xor(s, msk, 32);
            const int   oi = __shfl_xor(i, msk, 32);
            if (os < s || (os == s && oi < i)) { s = os; i = oi; }
        }
        if (ml == 0) {
            const int row = mbase + r + (kb ? 8 : 0);  // lane0 -> r, lane16 -> r+8
            idx_out[row]   = i;
            idx_f_out[row] = (float)i;
        }
    }
}

// ---------------------------------------------------------------------------
// Fan-out: one block per row.
//  - one-hot row of 1024 floats, coalesced + non-temporal (write-once stream)
//  - z_q gather + straight-through value z + (z_q - z) back to NCHW (NT)
//  - deterministic per-row loss partial via LDS, fixed summation order
// ---------------------------------------------------------------------------
__global__ void __launch_bounds__(256) k_writeback(
        const float* __restrict__ z, const float* __restrict__ emb,
        const int* __restrict__ idx,
        float* __restrict__ out_zq, float* __restrict__ out_oh,
        float* __restrict__ partial)
{
    __shared__ float lds[64];
    const int row = blockIdx.x;
    const int t   = threadIdx.x;
    const int id  = idx[row];

    float* ohrow = out_oh + (size_t)row * NE;
    #pragma unroll
    for (int j = 0; j < 4; ++j) {
        const int col = t + j * 256;
        __builtin_nontemporal_store((col == id) ? 1.0f : 0.0f, ohrow + col);
    }

    if (t < EDIM) {
        const int bb = row >> 10, hw = row & 1023;
        const float zv = z[bb * CHW + t * HW + hw];
        const float zq = emb[id * EDIM + t];
        const float d  = zq - zv;
        // zp + (z_q - zp): the reference's straight-through float expression
        __builtin_nontemporal_store(zv + d, out_zq + bb * CHW + t * HW + hw);
        lds[t] = d * d;
    }
    __syncthreads();
    if (t == 0) {
        float s = 0.f;
        #pragma unroll
        for (int k = 0; k < EDIM; ++k) s += lds[k];
        partial[row] = s;
    }
}

// ---------------------------------------------------------------------------
// Final deterministic reduction: loss = 1.25 * sum / 2097152
// ---------------------------------------------------------------------------
__global__ void __launch_bounds__(256) k_reduce(
        const float* __restrict__ partial, float* __restrict__ out_loss)
{
    __shared__ float lds[256];
    const int t = threadIdx.x;
    float s = 0.f;
    for (int i = 0; i < NVEC / 256; ++i) s += partial[t + i * 256];
    lds[t] = s;
    __syncthreads();
    if (t == 0) {
        float tot = 0.f;
        for (int i = 0; i < 256; ++i) tot += lds[i];
        out_loss[0] = 1.25f * tot / 2097152.0f;
    }
}

// ---------------------------------------------------------------------------
extern "C" void kernel_launch(void* const* d_in, const int* in_sizes, int n_in,
                              void* d_out, int out_size, void* d_ws, size_t ws_size,
                              hipStream_t stream) {
    const float* z   = (const float*)d_in[0];   // [32,64,32,32]
    const float* emb = (const float*)d_in[1];   // [1024,64]
    float* out = (float*)d_out;

    // d_out layout (concat, return order): loss | z_q(NCHW) | one_hot | idx
    float* out_loss = out;
    float* out_zq   = out + 1;
    float* out_oh   = out + 1 + (size_t)TOT;
    float* out_idxf = out + 1 + (size_t)TOT + (size_t)NVEC * NE;

    // workspace: enorm(4KB) | idx(128KB) | partial(128KB)  (~260KB total)
    char*  ws      = (char*)d_ws;
    float* enorm   = (float*)ws;
    int*   idx     = (int*)(ws + 4096);
    float* partial = (float*)(ws + 4096 + (size_t)NVEC * sizeof(int));

    k_enorm<<<(NE + 255) / 256, 256, 0, stream>>>(emb, enorm);
    // 2048 M-tiles, 4 waves (128 threads) per block -> 512 blocks
    k_gemm_argmin<<<(NVEC / 16) / 4, 128, 0, stream>>>(z, emb, enorm, idx, out_idxf);
    k_writeback<<<NVEC, 256, 0, stream>>>(z, emb, idx, out_zq, out_oh, partial);
    k_reduce<<<1, 256, 0, stream>>>(partial, out_loss);
}